// CTCLossLayer_79559974191532
// MI455X (gfx1250) — compile-verified
//
#include <hip/hip_runtime.h>
#include <stdint.h>

// CTC forward (K.ctc_batch_cost) for B=1024, T=256, C=100 (blank=99), L=64.
// One wave32 per batch element; scaled linear-space forward DP.
// gfx1250 async Global->LDS staging: 6-deep pipeline over an 8-row LDS ring,
// synchronized with s_wait_asynccnt (in-order async completion).

#define Bn    1024
#define Tn    256
#define Cn    100
#define Ln    64
#define Sn    129          // 2*L+1 lattice states
#define BLANK 99
#define EPSI  1e-7f
#define KPL   5            // states per lane: 32*5 = 160 >= 129
#define ROWPAD 112         // padded LDS row stride (floats); 448 B, 16B aligned
#define NBUF  8            // LDS ring size (power of 2)
#define DEPTH 6            // outstanding async stages

__global__ __launch_bounds__(32)
void ctc_fwd_kernel(const int* __restrict__ y_true,
                    const float* __restrict__ y_pred,
                    float* __restrict__ out)
{
    __shared__ __align__(16) float smem[NBUF * ROWPAD];
    const int lane = threadIdx.x;   // 0..31 (wave32)
    const int b    = blockIdx.x;    // batch element

    // ---- per-state constants (registers) ----
    int   cls[KPL];
    float allowf[KPL], validf[KPL];
#pragma unroll
    for (int k = 0; k < KPL; ++k) {
        int  s   = lane * KPL + k;
        bool odd = (s & 1) != 0;
        int  li  = (s - 1) >> 1;
        if (li < 0) li = 0;
        if (li > Ln - 1) li = Ln - 1;
        int lab  = y_true[b * Ln + li];
        int labp = y_true[b * Ln + (li > 0 ? li - 1 : 0)];
        cls[k]    = odd ? lab : BLANK;
        allowf[k] = (odd && (s == 1 || lab != labp)) ? 1.0f : 0.0f; // ext[s]!=ext[s-2]
        validf[k] = (s < Sn) ? 1.0f : 0.0f;
    }

    // low 32 bits of a __shared__ flat address == LDS allocation offset
    const uint32_t lds0 = (uint32_t)(uintptr_t)(void*)&smem[0];

    // stage row t into ring slot (t & 7): 25 lanes x b128 = 400 bytes.
    // s_wait_dscnt 0 first: prior iterations' ds_load gathers must retire
    // before the async engine overwrites their (recycled) slot (ASYNCcnt and
    // DScnt are unordered w.r.t. each other).
    auto stage = [&](int t) {
        asm volatile("s_wait_dscnt 0" ::: "memory");
        if (lane < 25) {
            uint32_t voff = (uint32_t)(((unsigned)b * Tn + (unsigned)t) * (Cn * 4))
                          + (uint32_t)lane * 16u;
            uint32_t la   = lds0 + (uint32_t)((t & (NBUF - 1)) * ROWPAD * 4)
                          + (uint32_t)lane * 16u;
            asm volatile("global_load_async_to_lds_b128 %0, %1, %2 offset:0"
                         :: "v"(la), "v"(voff), "s"(y_pred)
                         : "memory");
        }
    };
#define AWAIT_OLDEST() asm volatile("s_wait_asynccnt 5" ::: "memory")
#define AWAIT_ALL()    asm volatile("s_wait_asynccnt 0" ::: "memory")

    // prologue: fill pipeline with rows 0..DEPTH-1
#pragma unroll
    for (int i = 0; i < DEPTH; ++i) stage(i);

    float a[KPL];
    float logL = 0.0f;

    // ---- t = 0: only states 0 and 1 reachable ----
    AWAIT_OLDEST();           // issued=6, cnt<=5 -> row 0 complete (in-order)
    stage(DEPTH);             // keep 6 in flight
    {
        const float* row = &smem[0];
#pragma unroll
        for (int k = 0; k < KPL; ++k) {
            int   s = lane * KPL + k;
            float p = row[cls[k]] + EPSI;
            a[k] = (s < 2) ? p : 0.0f;
        }
    }

    // ---- time recurrence; renormalize every 4 steps (exact rescaling) ----
    for (int t = 1; t < Tn; ++t) {
        if (t + DEPTH < Tn) { AWAIT_OLDEST(); stage(t + DEPTH); }
        else                { AWAIT_ALL(); }
        const float* row = &smem[(t & (NBUF - 1)) * ROWPAD];

        // neighbor states from previous lane (s-1 / s-2 at lane boundary)
        float up4 = __shfl_up(a[KPL - 1], 1, 32);  // state 5l-1
        float up3 = __shfl_up(a[KPL - 2], 1, 32);  // state 5l-2
        if (lane == 0) { up4 = 0.0f; up3 = 0.0f; } // no predecessors for s<2

        float na[KPL];
#pragma unroll
        for (int k = 0; k < KPL; ++k) {
            float am1 = (k == 0) ? up4 : a[k - 1];
            float am2 = (k == 0) ? up3 : (k == 1) ? up4 : a[k - 2];
            float p   = row[cls[k]] + EPSI;
            na[k] = (a[k] + am1 + allowf[k] * am2) * p * validf[k];
        }

        if ((t & 3) == 3) {
            // worst-case shrink over 4 steps is eps^4 = 1e-28 >> fp32 min-normal,
            // so renormalizing every 4th step is numerically safe.
            float r = na[0] + na[1] + na[2] + na[3] + na[4];
#pragma unroll
            for (int off = 16; off > 0; off >>= 1) r += __shfl_xor(r, off, 32);
            float inv = 1.0f / r;
            logL += __logf(r);
#pragma unroll
            for (int k = 0; k < KPL; ++k) a[k] = na[k] * inv;
        } else {
#pragma unroll
            for (int k = 0; k < KPL; ++k) a[k] = na[k];
        }
    }

    // loss = -(logL + log(alpha[S-1] + alpha[S-2])); states 127,128 -> lane 25 k=2,3
    float v127 = __shfl(a[2], 25, 32);
    float v128 = __shfl(a[3], 25, 32);
    if (lane == 0) out[b] = -(logL + __logf(v127 + v128));
}

extern "C" void kernel_launch(void* const* d_in, const int* in_sizes, int n_in,
                              void* d_out, int out_size, void* d_ws, size_t ws_size,
                              hipStream_t stream) {
    const int*   y_true = (const int*)d_in[0];   // [B, L] int32
    const float* y_pred = (const float*)d_in[1]; // [B, T, C] float32 (probabilities)
    float*       outp   = (float*)d_out;         // [B, 1] float32
    (void)in_sizes; (void)n_in; (void)out_size; (void)d_ws; (void)ws_size;
    hipLaunchKernelGGL(ctc_fwd_kernel, dim3(Bn), dim3(32), 0, stream,
                       y_true, y_pred, outp);
}